// DigitCaps_54365696033038
// MI455X (gfx1250) — compile-verified
//
#include <hip/hip_runtime.h>
#include <hip/hip_bf16.h>
#include <math.h>

typedef __attribute__((ext_vector_type(2))) float v2f;
typedef __attribute__((ext_vector_type(8))) float v8f;

#define CC    32
#define INC   8
#define OC    10
#define PC    16
#define HWN   144
#define NN    (CC*HWN)        // 4608
#define BLOCK 512
#define NWAVE 16
#define XPB   (NN*INC)        // 36864 floats / batch
#define PADA  20              // padded row pitch for per-wave a/c scratch

// dynamic LDS layout (floats)
#define OFF_U    0
#define OFF_RED  (OFF_U + XPB)                  // 16 waves * 160
#define OFF_S    (OFF_RED + NWAVE*OC*PC)
#define OFF_V0   (OFF_S + OC*PC)
#define OFF_VR   (OFF_V0 + OC*PC)
#define OFF_USUM (OFF_VR + OC*PC)               // 32*8
#define OFF_FAC  (OFF_USUM + CC*INC)
#define OFF_WV   (OFF_FAC + 16)                 // 32*16*8 (o padded to 16, o>=10 zeroed)
#define OFF_A    (OFF_WV + CC*16*INC)           // per-wave 16x20 scratch
#define SMEM_FLOATS (OFF_A + NWAVE*16*PADA)     // 49392 floats = 197568 B

#define WMMA_F32(A, B, C) \
  __builtin_amdgcn_wmma_f32_16x16x4_f32(false, (A), false, (B), (short)0, (C), false, false)

__global__ __launch_bounds__(BLOCK)
void digitcaps_routing_kernel(const float* __restrict__ x,
                              const float* __restrict__ Wg,
                              float* __restrict__ out) {
  extern __shared__ float sm[];
  float* sU   = sm + OFF_U;     // x[b]: [c][hw][q]
  float* sRed = sm + OFF_RED;
  float* sS   = sm + OFF_S;
  float* sV0  = sm + OFF_V0;
  float* sVr  = sm + OFF_VR;
  float* usum = sm + OFF_USUM;
  float* sFac = sm + OFF_FAC;
  float* sWv  = sm + OFF_WV;    // Wv[c][o16][q]
  float* sA   = sm + OFF_A;     // per-wave [16 n][PADA] logits -> couplings

  const int tid  = threadIdx.x;
  const int lane = tid & 31;
  const int wave = tid >> 5;
  const int row  = lane & 15;
  const int hi   = lane >> 4;
  const int b    = blockIdx.x;

  // ---- stage x[b] into LDS ----
  {
    const float4* xg = (const float4*)(x + (size_t)b * XPB);
    float4* ul = (float4*)sU;
    #pragma unroll 2
    for (int i = tid; i < XPB/4; i += BLOCK) ul[i] = xg[i];
  }
  __syncthreads();

  // ================= iteration 0: uniform coupling 1/10 =================
  // s0[o,p] = 0.1 * sum_c sum_q W[c,o,p,q] * (sum_hw u[c,hw,q])
  if (tid < CC*INC) {
    int c = tid >> 3, q = tid & 7;
    float s = 0.f;
    const float* up = sU + c*HWN*INC + q;
    for (int hw = 0; hw < HWN; ++hw) s += up[hw*INC];
    usum[tid] = s;
  }
  __syncthreads();
  if (tid < OC*PC) {
    int o = tid >> 4, p = tid & 15;
    float s = 0.f;
    for (int c = 0; c < CC; ++c) {
      const float* wb = Wg + ((c*OC + o)*PC + p)*INC;
      const float4 w0 = *(const float4*)(wb);
      const float4 w1 = *(const float4*)(wb + 4);
      const float* us = usum + c*INC;
      s += w0.x*us[0] + w0.y*us[1] + w0.z*us[2] + w0.w*us[3]
         + w1.x*us[4] + w1.y*us[5] + w1.z*us[6] + w1.w*us[7];
    }
    sS[tid] = 0.1f * s;
  }
  __syncthreads();
  if (tid < OC) {
    float sn = 0.f;
    #pragma unroll
    for (int p = 0; p < PC; ++p) { float t = sS[tid*PC + p]; sn += t*t; }
    sFac[tid] = sn / ((1.f + sn) * sqrtf(sn));
  }
  __syncthreads();
  if (tid < OC*PC) {
    float v = sS[tid] * sFac[tid >> 4];
    sV0[tid] = v;
    sVr[tid] = v;
  }
  __syncthreads();

  // ========= routing iterations 1,2 (pass1 uses v0, pass2 uses v0+v1) =========
  for (int pass = 0; pass < 2; ++pass) {
    // ---- Wv[c,o,q] = sum_p W[c,o,p,q] * vr[o,p]; one (c,o16) slot per thread ----
    {
      int c = tid >> 4, o = tid & 15;
      float wv[INC];
      #pragma unroll
      for (int q = 0; q < INC; ++q) wv[q] = 0.f;
      if (o < OC) {
        const float* wb = Wg + (size_t)(c*OC + o)*PC*INC;
        #pragma unroll
        for (int p = 0; p < PC; ++p) {
          float vp = sVr[o*PC + p];
          const float4 w0 = *(const float4*)(wb + p*INC);
          const float4 w1 = *(const float4*)(wb + p*INC + 4);
          wv[0] += w0.x*vp; wv[1] += w0.y*vp; wv[2] += w0.z*vp; wv[3] += w0.w*vp;
          wv[4] += w1.x*vp; wv[5] += w1.y*vp; wv[6] += w1.z*vp; wv[7] += w1.w*vp;
        }
      }
      float4* dst = (float4*)(sWv + tid*INC);   // (c*16+o)*8 == tid*8
      dst[0] = float4{wv[0], wv[1], wv[2], wv[3]};
      dst[1] = float4{wv[4], wv[5], wv[6], wv[7]};
    }
    __syncthreads();

    float sP[PC];                     // s[o=row][p], accumulated over this wave's channels
    #pragma unroll
    for (int p = 0; p < PC; ++p) sP[p] = 0.f;

    for (int ci = 0; ci < 2; ++ci) {
      const int c = wave + ci*NWAVE;  // 2 channels per wave, 9 tiles each
      v8f G = {};                     // G[q=0..7 rows][o cols]; accumulates via C operand
      float* aw = sA + wave*(16*PADA);

      for (int t9 = 0; t9 < 9; ++t9) {
        const float* up = sU + (c*HWN + t9*16)*INC;

        // ---- agreement logits: a[16n x 16o] = u(16x8) x Wv^T(8x16) ----
        v2f A0 = *(const v2f*)(up + row*INC + 2*hi);       // u[n=row][q=2hi+{0,1}]
        v2f A1 = *(const v2f*)(up + row*INC + 4 + 2*hi);
        const float* wvp = sWv + (c*16 + row)*INC + 2*hi;  // Wv[c][o=row][q]
        v2f B0 = *(const v2f*)(wvp);
        v2f B1 = *(const v2f*)(wvp + 4);
        v8f Aacc = {};
        Aacc = WMMA_F32(A0, B0, Aacc);
        Aacc = WMMA_F32(A1, B1, Aacc);

        // ---- spill logits to per-wave scratch: aw[n][o] ----
        #pragma unroll
        for (int r = 0; r < 8; ++r) aw[(r + 8*hi)*PADA + row] = Aacc[r];

        // ---- softmax over o (hi==0 lanes, one n-row each), write c in place ----
        if (hi == 0) {
          float* ar = aw + row*PADA;
          float4 a0 = *(float4*)(ar);
          float4 a1 = *(float4*)(ar + 4);
          v2f    a2 = *(v2f*)(ar + 8);
          float mx = fmaxf(fmaxf(fmaxf(a0.x, a0.y), fmaxf(a0.z, a0.w)),
                     fmaxf(fmaxf(fmaxf(a1.x, a1.y), fmaxf(a1.z, a1.w)),
                           fmaxf(a2.x, a2.y)));
          float e0 = __expf(a0.x - mx), e1 = __expf(a0.y - mx);
          float e2 = __expf(a0.z - mx), e3 = __expf(a0.w - mx);
          float e4 = __expf(a1.x - mx), e5 = __expf(a1.y - mx);
          float e6 = __expf(a1.z - mx), e7 = __expf(a1.w - mx);
          float e8 = __expf(a2.x - mx), e9 = __expf(a2.y - mx);
          float inv = 1.0f / (e0+e1+e2+e3+e4+e5+e6+e7+e8+e9);
          *(float4*)(ar)     = float4{e0*inv, e1*inv, e2*inv, e3*inv};
          *(float4*)(ar + 4) = float4{e4*inv, e5*inv, e6*inv, e7*inv};
          *(v2f*)(ar + 8)    = v2f{e8*inv, e9*inv};
        }

        // ---- G[q,o] += u^T(8x16n) x c(16n x 16o); K = n in 4 steps ----
        #pragma unroll
        for (int s4 = 0; s4 < 4; ++s4) {
          const int n0 = 4*s4 + 2*hi;
          v2f Au; Au.x = up[n0*INC + row];        // u[n][q=row]  (rows q>=8 garbage, discarded)
                  Au.y = up[(n0+1)*INC + row];
          v2f Bc; Bc.x = aw[n0*PADA + row];       // c[n][o=row]  (cols o>=10 exactly 0)
                  Bc.y = aw[(n0+1)*PADA + row];
          G = WMMA_F32(Au, Bc, G);
        }
      }

      // ---- valid G lives in hi==0 lanes' regs; mirror to all lanes ----
      float Gq[8];
      #pragma unroll
      for (int r = 0; r < 8; ++r) {
        float go = __shfl_xor(G[r], 16, 32);
        Gq[r] = hi ? go : G[r];                   // Gq[r] = G[q=r][o=row]
      }
      // ---- s[o,p] += sum_q W[c,o,p,q] * G[q,o]; lane handles o=row ----
      const int osafe = (row < OC) ? row : 0;
      const float* wb = Wg + (size_t)(c*OC + osafe)*PC*INC;
      #pragma unroll
      for (int p = 0; p < PC; ++p) {
        const float4 w0 = *(const float4*)(wb + p*INC);
        const float4 w1 = *(const float4*)(wb + p*INC + 4);
        sP[p] += w0.x*Gq[0] + w0.y*Gq[1] + w0.z*Gq[2] + w0.w*Gq[3]
               + w1.x*Gq[4] + w1.y*Gq[5] + w1.z*Gq[6] + w1.w*Gq[7];
      }
    }

    // ---- block reduction of s over waves, then squash ----
    if (hi == 0 && row < OC) {
      float* sr = sRed + wave*(OC*PC) + row*PC;
      *(float4*)(sr)      = float4{sP[0],  sP[1],  sP[2],  sP[3]};
      *(float4*)(sr + 4)  = float4{sP[4],  sP[5],  sP[6],  sP[7]};
      *(float4*)(sr + 8)  = float4{sP[8],  sP[9],  sP[10], sP[11]};
      *(float4*)(sr + 12) = float4{sP[12], sP[13], sP[14], sP[15]};
    }
    __syncthreads();
    if (tid < OC*PC) {
      float s = 0.f;
      #pragma unroll
      for (int w = 0; w < NWAVE; ++w) s += sRed[w*(OC*PC) + tid];
      sS[tid] = s;
    }
    __syncthreads();
    if (tid < OC) {
      float sn = 0.f;
      #pragma unroll
      for (int p = 0; p < PC; ++p) { float t = sS[tid*PC + p]; sn += t*t; }
      sFac[tid] = sn / ((1.f + sn) * sqrtf(sn));
    }
    __syncthreads();
    if (pass == 0) {
      if (tid < OC*PC) sVr[tid] = sV0[tid] + sS[tid] * sFac[tid >> 4];  // v0 + v1
      __syncthreads();
    }
  }

  // final v2 = squash(s2)
  if (tid < OC*PC) out[(size_t)b*(OC*PC) + tid] = sS[tid] * sFac[tid >> 4];
}

extern "C" void kernel_launch(void* const* d_in, const int* in_sizes, int n_in,
                              void* d_out, int out_size, void* d_ws, size_t ws_size,
                              hipStream_t stream) {
  const float* x = (const float*)d_in[0];
  const float* W = (const float*)d_in[1];
  float* outp = (float*)d_out;

  const int Bn = in_sizes[0] / XPB;                 // 128
  const size_t smem = SMEM_FLOATS * sizeof(float);  // ~193 KB

  hipFuncSetAttribute((const void*)digitcaps_routing_kernel,
                      hipFuncAttributeMaxDynamicSharedMemorySize, (int)smem);
  digitcaps_routing_kernel<<<Bn, BLOCK, smem, stream>>>(x, W, outp);
}